// DGLGCNConv_21423296872969
// MI455X (gfx1250) — compile-verified
//
#include <hip/hip_runtime.h>

#define D 128

typedef __attribute__((ext_vector_type(2))) float v2f;
typedef __attribute__((ext_vector_type(8))) float v8f;
typedef __attribute__((ext_vector_type(4))) unsigned v4u;
typedef __attribute__((ext_vector_type(8))) unsigned v8u;

// ---------------------------------------------------------------- deg = 1.0
__global__ void fill_deg_kernel(float* __restrict__ deg, int N) {
    int i = blockIdx.x * blockDim.x + threadIdx.x;
    if (i < N) deg[i] = 1.0f;
}

// ------------------------------------------------- deg[dst[e]] += 1 (atomic)
__global__ void count_deg_kernel(const int* __restrict__ dst,
                                 float* __restrict__ deg, int E) {
    int e = blockIdx.x * blockDim.x + threadIdx.x;
    if (e < E) atomicAdd(&deg[dst[e]], 1.0f);
}

// ------------------------------------------------------- h = x @ W^T + b
// Block = 256 threads = 8 wave32, covering 128 rows of x. Wave w owns rows
// [blk*128 + 16w, +16) and computes ALL 8 column tiles (8 x v8f accumulators),
// so each A fragment is loaded from global once and reused by 8 WMMAs.
// W (64 KB) is staged row-major into LDS by the Tensor Data Mover (wave 0
// issues tensor_load_to_lds with a 2-group D# descriptor, waits TENSORcnt).
//
// V_WMMA_F32_16X16X4_F32 fragment layouts (§7.12.2):
//   A 16x4 : lane<16 -> K=0,1 in v0,v1 ; lane>=16 -> K=2,3
//   B 4x16 : symmetric (lane&15 = N column, lane>=16 -> K=2,3)
//   C/D    : VGPR r -> M = r + 8*(lane>=16), N = lane&15
__global__ __launch_bounds__(256) void gemm_xwt_wmma(
        const float* __restrict__ x, const float* __restrict__ W,
        const float* __restrict__ bvec, float* __restrict__ h, int N) {
    __shared__ float Ws[D * D];               // row-major W, 64 KB

    const int tid  = threadIdx.x;
    const int wave = tid >> 5;
    const int lane = tid & 31;
    const int lo   = lane & 15;
    const int hi   = lane >> 4;

    if (wave == 0) {
        // ---- D# group 0 (128b): count=1 | lds_addr | global_addr | type=2
        const unsigned long long wa = (unsigned long long)(const void*)W;
        const unsigned lds_base = (unsigned)(unsigned long long)(void*)Ws;
        v4u g0;
        g0.x = 1u;                                        // count=1, user mode
        g0.y = lds_base;                                  // LDS byte address
        g0.z = (unsigned)(wa & 0xFFFFFFFFull);            // global_addr[31:0]
        g0.w = (unsigned)((wa >> 32) & 0x01FFFFFFull)     // global_addr[56:32]
             | (2u << 30);                                // type = 2 (image)
        // ---- D# group 1 (256b): data_size=4B, 128x128 tensor, full tile
        v8u g1;
        g1.s0 = (2u << 16);                               // data_size = 4 bytes
        g1.s1 = (128u << 16);                             // tensor_dim0 = 128 (lo16)
        g1.s2 = (128u << 16);                             // tensor_dim1 = 128 (lo16)
        g1.s3 = (128u << 16);                             // tile_dim0 = 128
        g1.s4 = 128u;                                     // tile_dim1 = 128, tile_dim2 = 0
        g1.s5 = 128u;                                     // tensor_dim0_stride = 128 (lo32)
        g1.s6 = 0u;                                       // stride hi / dim1_stride lo
        g1.s7 = 0u;                                       // dim1_stride hi
        asm volatile("tensor_load_to_lds %0, %1" :: "s"(g0), "s"(g1) : "memory");
        __builtin_amdgcn_s_wait_tensorcnt(0);
    }
    __syncthreads();

    const int mbase = blockIdx.x * 128 + wave * 16;
    int mrow = mbase + lo;
    if (mrow >= N) mrow = N - 1;              // clamp loads; EXEC all-1s for WMMA

    // 8 accumulators, seeded with bias (depends only on column).
    v8f acc[8];
    #pragma unroll
    for (int j = 0; j < 8; ++j) {
        const float bias = bvec[j * 16 + lo];
        #pragma unroll
        for (int r = 0; r < 8; ++r) acc[j][r] = bias;
    }

    const float* __restrict__ xrow = x + (size_t)mrow * D;

    for (int k = 0; k < D; k += 4) {
        v2f a;
        a.x = xrow[k + 2 * hi + 0];                       // one global b64 / lane
        a.y = xrow[k + 2 * hi + 1];
        #pragma unroll
        for (int j = 0; j < 8; ++j) {
            // B[kk][n] = W[n][kk]; consecutive kk -> single aligned ds b64
            v2f bb = *(const v2f*)&Ws[(j * 16 + lo) * D + k + 2 * hi];
            acc[j] = __builtin_amdgcn_wmma_f32_16x16x4_f32(
                         false, a, false, bb, (short)0, acc[j], false, false);
        }
    }

    if (mbase + 16 <= N) {                    // wave-uniform fast path
        #pragma unroll
        for (int j = 0; j < 8; ++j) {
            float* hp = h + (size_t)(mbase + 8 * hi) * D + j * 16 + lo;
            #pragma unroll
            for (int r = 0; r < 8; ++r) hp[(size_t)r * D] = acc[j][r];
        }
    } else {
        #pragma unroll
        for (int j = 0; j < 8; ++j) {
            #pragma unroll
            for (int r = 0; r < 8; ++r) {
                int row = mbase + r + 8 * hi;
                if (row < N) h[(size_t)row * D + j * 16 + lo] = acc[j][r];
            }
        }
    }
}

// ------------------------- out = relu(h + root_emb) * (1/deg)   (float4 wide)
__global__ void self_loop_kernel(const float* __restrict__ h,
                                 const float* __restrict__ root,
                                 const float* __restrict__ deg,
                                 float* __restrict__ out, int nquads) {
    int idx = blockIdx.x * blockDim.x + threadIdx.x;     // one float4 per thread
    if (idx >= nquads) return;
    const int n = idx >> 5;                               // 32 quads per node
    const int q = idx & 31;
    float4 hv = ((const float4*)h)[idx];
    float4 rv = ((const float4*)root)[q];
    const float inv = 1.0f / deg[n];
    float4 o;
    o.x = hv.x + rv.x; o.x = (o.x > 0.0f ? o.x : 0.0f) * inv;
    o.y = hv.y + rv.y; o.y = (o.y > 0.0f ? o.y : 0.0f) * inv;
    o.z = hv.z + rv.z; o.z = (o.z > 0.0f ? o.z : 0.0f) * inv;
    o.w = hv.w + rv.w; o.w = (o.w > 0.0f ? o.w : 0.0f) * inv;
    ((float4*)out)[idx] = o;
}

// -------------------- out[dst] += norm * relu(h[src])  (one wave per edge)
// Lane l gathers float4 h[src*128 + 4l..4l+3] (contiguous 512 B per wave),
// then 4 fp32 atomics into out[dst] (25.6 MB output -> L2-resident atomics).
__global__ __launch_bounds__(256) void edge_scatter_kernel(
        const int* __restrict__ src, const int* __restrict__ dst,
        const float* __restrict__ h, const float* __restrict__ deg,
        float* __restrict__ out, int E) {
    const int lane = threadIdx.x & 31;
    const int wave = threadIdx.x >> 5;
    const int e = blockIdx.x * 8 + wave;
    if (e >= E) return;                       // wave-uniform exit

    const int s = src[e];
    const int t = dst[e];
    const float norm = rsqrtf(deg[s]) * rsqrtf(deg[t]);

    const float4* __restrict__ hs = (const float4*)(h + (size_t)s * D);
    float4 v = hs[lane];
    v.x = v.x > 0.0f ? v.x * norm : 0.0f;
    v.y = v.y > 0.0f ? v.y * norm : 0.0f;
    v.z = v.z > 0.0f ? v.z * norm : 0.0f;
    v.w = v.w > 0.0f ? v.w * norm : 0.0f;

    float* o = out + (size_t)t * D + lane * 4;
    atomicAdd(o + 0, v.x);
    atomicAdd(o + 1, v.y);
    atomicAdd(o + 2, v.z);
    atomicAdd(o + 3, v.w);
}

extern "C" void kernel_launch(void* const* d_in, const int* in_sizes, int n_in,
                              void* d_out, int out_size, void* d_ws, size_t ws_size,
                              hipStream_t stream) {
    const float* x    = (const float*)d_in[0];
    const int*   src  = (const int*)d_in[1];
    const int*   dst  = (const int*)d_in[2];
    const float* W    = (const float*)d_in[3];
    const float* b    = (const float*)d_in[4];
    const float* root = (const float*)d_in[5];

    const int N = in_sizes[0] / D;
    const int E = in_sizes[1];

    float* out = (float*)d_out;
    float* h   = (float*)d_ws;                 // N*D floats
    float* deg = h + (size_t)N * D;            // N floats

    fill_deg_kernel<<<(N + 255) / 256, 256, 0, stream>>>(deg, N);
    count_deg_kernel<<<(E + 255) / 256, 256, 0, stream>>>(dst, deg, E);
    gemm_xwt_wmma<<<(N + 127) / 128, 256, 0, stream>>>(x, W, b, h, N);
    const int nquads = N * (D / 4);
    self_loop_kernel<<<(nquads + 255) / 256, 256, 0, stream>>>(h, root, deg, out, nquads);
    edge_scatter_kernel<<<(E + 7) / 8, 256, 0, stream>>>(src, dst, h, deg, out, E);
}